// VoxelQueryAndGrouping_781684048001
// MI455X (gfx1250) — compile-verified
//
#include <hip/hip_runtime.h>
#include <stdint.h>

// Voxel query + grouping for MI455X (gfx1250, wave32).
// One wave32 per query; 8 queries per 256-thread block.
// CDNA5 paths used: wave32 ballot compaction, GLOBAL_LOAD_ASYNC_TO_LDS_B128
// gather into LDS (ASYNCcnt) overlapped with xyz output, s_wait_asynccnt,
// conflict-free LDS transpose, b128 coalesced stores.

#define ZG 21
#define YG 400
#define XG 352
#define NS 16
#define CC 32
#define R2 64.0f
#define QPB 8 // queries (waves) per block

__global__ __launch_bounds__(256) void voxel_query_group_kernel(
    const int* __restrict__ new_coords,   // (M,4) int32 [b,z,y,x]
    const float* __restrict__ xyz,        // (N,3)
    const float* __restrict__ new_xyz,    // (M,3)
    const float* __restrict__ features,   // (N,32)
    const int* __restrict__ v2p,          // (B,Z,Y,X)
    const int* __restrict__ xbc,          // (B,) xyz_batch_cnt
    int M,
    float* __restrict__ out_f,            // (M,32,16)
    float* __restrict__ out_x,            // (M,3,16)
    float* __restrict__ out_e)            // (M,)
{
    __shared__ int   s_idx[QPB][NS];
    __shared__ float s_feat[QPB][NS][CC]; // 16 rows x 128B per wave

    const int lane = threadIdx.x & 31;
    const int wave = threadIdx.x >> 5;
    const int m    = blockIdx.x * QPB + wave;
    const bool active = (m < M); // wave-uniform

    int b = 0, zc = 0, yc = 0, xc = 0;
    float qx = 0.f, qy = 0.f, qz = 0.f;
    int cnt = 0;

    if (active) {
        const int4 cd = ((const int4*)new_coords)[m];
        b = cd.x; zc = cd.y; yc = cd.z; xc = cd.w;
        qx = new_xyz[(size_t)m * 3 + 0];
        qy = new_xyz[(size_t)m * 3 + 1];
        qz = new_xyz[(size_t)m * 3 + 2];

        // ---- Phase 1: ordered neighborhood scan, 32 candidates per step ----
        for (int k = 0; k < 4; ++k) {
            int cand = k * 32 + lane;       // scan order: dz major, dy, dx minor
            bool valid = false;
            int pidx = -1;
            if (cand < 125) {
                int dz = cand / 25 - 2;
                int dy = (cand / 5) % 5 - 2;
                int dx = cand % 5 - 2;
                int z = zc + dz, y = yc + dy, x = xc + dx;
                if (z >= 0 && z < ZG && y >= 0 && y < YG && x >= 0 && x < XG) {
                    pidx = v2p[(((size_t)b * ZG + z) * YG + y) * XG + x];
                    if (pidx >= 0) {
                        float px = xyz[(size_t)pidx * 3 + 0] - qx;
                        float py = xyz[(size_t)pidx * 3 + 1] - qy;
                        float pz = xyz[(size_t)pidx * 3 + 2] - qz;
                        valid = (px * px + py * py + pz * pz) < R2;
                    }
                }
            }
            // wave32 ballot + prefix popcount -> ordered slot assignment
            unsigned mask = __builtin_amdgcn_ballot_w32(valid);
            int slot = cnt + __popc(mask & ((1u << lane) - 1u));
            if (valid && slot < NS) s_idx[wave][slot] = pidx;
            cnt += __popc(mask);              // uniform across wave
            if (cnt >= NS) break;             // uniform branch
        }
    }
    __syncthreads();

    // ---- Finalize indices: tail-fill with first pick, or batch-offset if empty
    if (active) {
        if (lane < NS) {
            int g;
            if (cnt == 0) {
                int off = 0;
                for (int i = 0; i < b; ++i) off += xbc[i];
                g = off;                      // gidx = xyz_offs[b] when empty
            } else {
                int total = cnt < NS ? cnt : NS;
                int first = s_idx[wave][0];
                g = (lane < total) ? s_idx[wave][lane] : first;
            }
            s_idx[wave][lane] = g;
        }
        if (lane == 0) out_e[m] = (cnt == 0) ? 1.0f : 0.0f;
    }
    __syncthreads();

    if (active) {
        // ---- Phase 2a: async DMA gather of 16 feature rows (2KB) into LDS.
        // Each lane moves 16B; 4 issues cover 16 rows x 128B. Tracked by ASYNCcnt.
        const uint32_t lds_base = (uint32_t)(uintptr_t)(&s_feat[wave][0][0]);
        const uint64_t fbase = (uint64_t)(uintptr_t)features;
#pragma unroll
        for (int k = 0; k < 4; ++k) {
            int li = k * 32 + lane;           // 0..127
            int s  = li >> 3;                 // row 0..15
            int j  = li & 7;                  // 16B chunk within the 128B row
            int g  = s_idx[wave][s];
            uint64_t ga = fbase + (uint64_t)(uint32_t)g * (CC * 4) + (uint32_t)(j * 16);
            uint32_t la = lds_base + (uint32_t)(li * 16);
            asm volatile("global_load_async_to_lds_b128 %0, %1, off"
                         :: "v"(la), "v"(ga) : "memory");
        }

        // ---- Phase 2b: grouped_xyz output while the async copies are in flight
        if (lane < NS) {
            int g = s_idx[wave][lane];
            float px = xyz[(size_t)g * 3 + 0];
            float py = xyz[(size_t)g * 3 + 1];
            float pz = xyz[(size_t)g * 3 + 2];
            size_t base = (size_t)m * (3 * NS);
            out_x[base + lane]          = px;   // d=0 row
            out_x[base + NS + lane]     = py;   // d=1 row
            out_x[base + 2 * NS + lane] = pz;   // d=2 row
        }

        // ---- Phase 2c: wait for DMA, transpose out of LDS, coalesced b128 stores
        asm volatile("s_wait_asynccnt 0" ::: "memory");

        const int c = lane;                   // channel per lane
        float f[NS];
#pragma unroll
        for (int s = 0; s < NS; ++s)
            f[s] = s_feat[wave][s][c];        // bank = (32*s + c) % 64: conflict-free

        float4* d4 = (float4*)(out_f + (size_t)m * (CC * NS) + (size_t)c * NS);
        d4[0] = make_float4(f[0],  f[1],  f[2],  f[3]);
        d4[1] = make_float4(f[4],  f[5],  f[6],  f[7]);
        d4[2] = make_float4(f[8],  f[9],  f[10], f[11]);
        d4[3] = make_float4(f[12], f[13], f[14], f[15]);
    }
}

extern "C" void kernel_launch(void* const* d_in, const int* in_sizes, int n_in,
                              void* d_out, int out_size, void* d_ws, size_t ws_size,
                              hipStream_t stream) {
    const int*   new_coords = (const int*)d_in[0];   // (M,4)
    const float* xyz        = (const float*)d_in[1]; // (N,3)
    const int*   xbc        = (const int*)d_in[2];   // (B,)
    const float* new_xyz    = (const float*)d_in[3]; // (M,3)
    // d_in[4] = new_xyz_batch_cnt (unused)
    const float* features   = (const float*)d_in[5]; // (N,32)
    const int*   v2p        = (const int*)d_in[6];   // (B,Z,Y,X)

    const int M = in_sizes[0] / 4;

    float* out_f = (float*)d_out;                       // (M,32,16)
    float* out_x = out_f + (size_t)M * CC * NS;         // (M,3,16)
    float* out_e = out_x + (size_t)M * 3 * NS;          // (M,)

    const int blocks = (M + QPB - 1) / QPB;
    voxel_query_group_kernel<<<blocks, 256, 0, stream>>>(
        new_coords, xyz, new_xyz, features, v2p, xbc, M, out_f, out_x, out_e);
}